// lstm_16707422782055
// MI455X (gfx1250) — compile-verified
//
#include <hip/hip_runtime.h>
#include <hip/hip_bf16.h>

// SRU (3-layer) for MI455X / gfx1250, fp32 throughout.
// GEMMs via V_WMMA_F32_16X16X4_F32: one wave computes a 64x16 C strip
// (4 M-tiles sharing each B fragment). Layer-0 inputs are pre-packed
// ((t,b)-ordered, K padded 37->40) so the GEMM inner loop is branch-free.
// Sequential scans as VALU kernels; head GEMV via wave-shuffle reduction.

typedef __attribute__((ext_vector_type(2))) float v2f;
typedef __attribute__((ext_vector_type(8))) float v8f;

#define HDIM 128
#define BDIM 128
#define TDIM 2048
#define FDIM 37
#define KPAD 40   // FDIM padded to a multiple of 4

// ---------------------------------------------------------------------------
// Pack x (B,T,F) -> xp (T*B rows, KPAD cols), row m = t*B + b, zero-padded.
// ---------------------------------------------------------------------------
__global__ void pack_x_pad(const float* __restrict__ x, float* __restrict__ xp) {
    const int idx = blockIdx.x * blockDim.x + threadIdx.x;
    const int total = TDIM * BDIM * KPAD;
    if (idx >= total) return;
    const int m = idx / KPAD;
    const int f = idx - m * KPAD;
    const int t = m >> 7;            // m / BDIM
    const int b = m & 127;           // m % BDIM
    float v = 0.f;
    if (f < FDIM) v = x[((size_t)b * TDIM + (size_t)t) * FDIM + f];
    xp[idx] = v;
}

// Zero-pad W0 (F x 4H) -> W0p (KPAD x 4H).
__global__ void pack_w0_pad(const float* __restrict__ W0, float* __restrict__ W0p) {
    const int N = 4 * HDIM;
    const int idx = blockIdx.x * blockDim.x + threadIdx.x;
    if (idx >= KPAD * N) return;
    const int k = idx / N;
    const int n = idx - k * N;
    W0p[idx] = (k < FDIM) ? W0[k * N + n] : 0.f;
}

// ---------------------------------------------------------------------------
// GEMM: C[M x N] = A[M x K] @ B[K x N], fp32. One wave per 64x16 C strip:
// 4 M-tiles share each B fragment (4 back-to-back WMMAs per k-step).
// M % 64 == 0, N % 16 == 0, K % 4 == 0. Branch-free inner loop.
// ---------------------------------------------------------------------------
__global__ void gemm_wmma_f32(const float* __restrict__ A,
                              const float* __restrict__ Bm,
                              float* __restrict__ C,
                              int M, int N, int K) {
    const int wave = blockIdx.x * (blockDim.x >> 5) + (threadIdx.x >> 5);
    const int lane = threadIdx.x & 31;
    const int tilesN  = N >> 4;
    const int stripsM = M >> 6;                 // 64 rows per strip
    const int strip = wave / tilesN;
    const int tileN = wave - strip * tilesN;
    if (strip >= stripsM) return;               // wave-uniform exit

    const int ml = lane & 15;                   // row (A) / col (B,C) in tile
    const int kh = lane >> 4;                   // K half-select

    const float* __restrict__ A0 = A + (size_t)(strip * 64 + ml) * (size_t)K;
    const float* __restrict__ A1 = A0 + (size_t)16 * (size_t)K;
    const float* __restrict__ A2 = A0 + (size_t)32 * (size_t)K;
    const float* __restrict__ A3 = A0 + (size_t)48 * (size_t)K;
    const float* __restrict__ Bcol = Bm + (tileN * 16 + ml);

    v8f c0 = {0.f,0.f,0.f,0.f,0.f,0.f,0.f,0.f};
    v8f c1 = c0, c2 = c0, c3 = c0;

    for (int k = 0; k < K; k += 4) {
        const int kk = k + 2 * kh;
        // B 4x16 fragment: lanes 0-15 -> K = k..k+1, lanes 16-31 -> K = k+2..k+3
        v2f bf;
        bf.x = Bcol[(size_t)(kk + 0) * (size_t)N];
        bf.y = Bcol[(size_t)(kk + 1) * (size_t)N];
        // A 16x4 fragments (float2 loads, 8B aligned)
        const v2f a0 = *(const v2f*)(A0 + kk);
        const v2f a1 = *(const v2f*)(A1 + kk);
        const v2f a2 = *(const v2f*)(A2 + kk);
        const v2f a3 = *(const v2f*)(A3 + kk);
        c0 = __builtin_amdgcn_wmma_f32_16x16x4_f32(false, a0, false, bf, (short)0, c0, false, false);
        c1 = __builtin_amdgcn_wmma_f32_16x16x4_f32(false, a1, false, bf, (short)0, c1, false, false);
        c2 = __builtin_amdgcn_wmma_f32_16x16x4_f32(false, a2, false, bf, (short)0, c2, false, false);
        c3 = __builtin_amdgcn_wmma_f32_16x16x4_f32(false, a3, false, bf, (short)0, c3, false, false);
    }

    // C/D layout: lane<16 -> rows r, lane>=16 -> rows r+8; col = ml
    float* __restrict__ Cp =
        C + (size_t)(strip * 64 + 8 * kh) * (size_t)N + (size_t)(tileN * 16 + ml);
    #pragma unroll
    for (int r = 0; r < 8; ++r) Cp[(size_t)r * (size_t)N] = c0[r];
    Cp += (size_t)16 * (size_t)N;
    #pragma unroll
    for (int r = 0; r < 8; ++r) Cp[(size_t)r * (size_t)N] = c1[r];
    Cp += (size_t)16 * (size_t)N;
    #pragma unroll
    for (int r = 0; r < 8; ++r) Cp[(size_t)r * (size_t)N] = c2[r];
    Cp += (size_t)16 * (size_t)N;
    #pragma unroll
    for (int r = 0; r < 8; ++r) Cp[(size_t)r * (size_t)N] = c3[r];
}

// ---------------------------------------------------------------------------
// SRU scan: one thread per (b, h). Sequential over t (f,r depend on running c).
//   f = sigmoid(fp + vf*c + bf); r = sigmoid(rp + vr*c + br)   [both use OLD c]
//   c = f*c + (1-f)*xt;          h = r*c + (1-r)*res
// For layer 0 pass res = U + 3*H with resStride = 4*H.
// ---------------------------------------------------------------------------
__global__ void sru_scan(const float* __restrict__ U, int Ustride,
                         const float* __restrict__ res, int resStride,
                         const float* __restrict__ vc,
                         const float* __restrict__ bias,
                         const float* __restrict__ c0,
                         float* __restrict__ hOut,
                         float* __restrict__ cOut) {
    const int idx = blockIdx.x * blockDim.x + threadIdx.x;
    if (idx >= BDIM * HDIM) return;
    const int b_i = idx >> 7;      // / HDIM
    const int hh  = idx & 127;     // % HDIM

    const float vf = vc[hh],   vr = vc[HDIM + hh];
    const float bf = bias[hh], br = bias[HDIM + hh];
    float c = c0[idx];

    const float* __restrict__ Up = U    + (size_t)b_i * (size_t)Ustride   + hh;
    const float* __restrict__ Rp = res  + (size_t)b_i * (size_t)resStride + hh;
    float*       __restrict__ Hp = hOut + (size_t)b_i * (size_t)HDIM      + hh;

    const size_t ustep = (size_t)BDIM * (size_t)Ustride;
    const size_t rstep = (size_t)BDIM * (size_t)resStride;
    const size_t hstep = (size_t)BDIM * (size_t)HDIM;

    for (int t = 0; t < TDIM; ++t) {
        const float xt = Up[0];
        const float fp = Up[HDIM];
        const float rp = Up[2 * HDIM];
        const float rs = Rp[0];
        const float f = 1.f / (1.f + __expf(-(fp + vf * c + bf)));
        const float r = 1.f / (1.f + __expf(-(rp + vr * c + br)));
        c = f * c + (1.f - f) * xt;
        Hp[0] = r * c + (1.f - r) * rs;
        Up += ustep; Rp += rstep; Hp += hstep;
    }
    cOut[idx] = c;
}

// ---------------------------------------------------------------------------
// Head GEMV: out[b*T + t, 0:4] = h[t*B + b, :] @ Wd + bd. One wave per row,
// coalesced loads of the h row, shuffle-tree reduction (wave32).
// ---------------------------------------------------------------------------
__global__ void head_gemv(const float* __restrict__ Hbuf,
                          const float* __restrict__ Wd,
                          const float* __restrict__ bd,
                          float* __restrict__ out) {
    const int wave = blockIdx.x * (blockDim.x >> 5) + (threadIdx.x >> 5);
    const int lane = threadIdx.x & 31;
    const int total = TDIM * BDIM;
    if (wave >= total) return;

    const int t   = wave / BDIM;
    const int b_i = wave - t * BDIM;
    const float* __restrict__ hrow = Hbuf + (size_t)wave * HDIM;

    float s0 = 0.f, s1 = 0.f, s2 = 0.f, s3 = 0.f;
    #pragma unroll
    for (int k = lane; k < HDIM; k += 32) {
        const float v = hrow[k];
        const float* w = Wd + k * 4;
        s0 += v * w[0]; s1 += v * w[1]; s2 += v * w[2]; s3 += v * w[3];
    }
    #pragma unroll
    for (int off = 16; off > 0; off >>= 1) {
        s0 += __shfl_down(s0, off, 32);
        s1 += __shfl_down(s1, off, 32);
        s2 += __shfl_down(s2, off, 32);
        s3 += __shfl_down(s3, off, 32);
    }
    if (lane == 0) {
        const size_t row = (size_t)b_i * (size_t)TDIM + (size_t)t;
        out[row * 4 + 0] = s0 + bd[0];
        out[row * 4 + 1] = s1 + bd[1];
        out[row * 4 + 2] = s2 + bd[2];
        out[row * 4 + 3] = s3 + bd[3];
    }
}

// ---------------------------------------------------------------------------
extern "C" void kernel_launch(void* const* d_in, const int* in_sizes, int n_in,
                              void* d_out, int out_size, void* d_ws, size_t ws_size,
                              hipStream_t stream) {
    (void)in_sizes; (void)n_in; (void)out_size; (void)ws_size;

    const float* x    = (const float*)d_in[0];   // (B, T, F)
    const float* c_in = (const float*)d_in[1];   // (L, B, H)
    const float* W0   = (const float*)d_in[2];   // (F, 4H)
    const float* W1   = (const float*)d_in[3];   // (H, 3H)
    const float* W2   = (const float*)d_in[4];   // (H, 3H)
    const float* vc0  = (const float*)d_in[5];
    const float* vc1  = (const float*)d_in[6];
    const float* vc2  = (const float*)d_in[7];
    const float* b0   = (const float*)d_in[8];
    const float* b1   = (const float*)d_in[9];
    const float* b2   = (const float*)d_in[10];
    const float* Wd   = (const float*)d_in[11];  // (H, 4)
    const float* bd   = (const float*)d_in[12];  // (4,)

    float* out = (float*)d_out;                  // [B*T*4] then [L*B*H]

    const int M = TDIM * BDIM;                   // 262144 GEMM rows
    // Workspace layout (floats):
    //   U   : M*512  (512 MB, reused for U0/U1/U2)
    //   H0  : M*128  (128 MB)
    //   H1  : M*128  (128 MB)
    //   xp  : M*KPAD ( 40 MB)
    //   W0p : KPAD*512
    float* U   = (float*)d_ws;
    float* H0  = U  + (size_t)M * 512;
    float* H1  = H0 + (size_t)M * HDIM;
    float* xp  = H1 + (size_t)M * HDIM;
    float* W0p = xp + (size_t)M * KPAD;

    float* cOutBase = out + (size_t)M * 4;       // c states appended after head out

    const dim3 blk256(256);
    const int scanGrid = (BDIM * HDIM + 255) / 256;

    // ---- Pack layer-0 operands (permute + zero-pad K to 40) ----
    pack_x_pad<<<(M * KPAD + 255) / 256, blk256, 0, stream>>>(x, xp);
    pack_w0_pad<<<(KPAD * 4 * HDIM + 255) / 256, blk256, 0, stream>>>(W0, W0p);

    // ---- Layer 0: U0 = xp @ W0p  (K = 40, N = 512) ----
    {
        const int N = 4 * HDIM, K = KPAD;
        const int waves = (M / 64) * (N / 16);
        gemm_wmma_f32<<<waves / 8, blk256, 0, stream>>>(xp, W0p, U, M, N, K);
    }
    sru_scan<<<scanGrid, blk256, 0, stream>>>(
        U, 4 * HDIM, /*res=*/U + 3 * HDIM, 4 * HDIM,
        vc0, b0, c_in + 0 * BDIM * HDIM, H0, cOutBase + 0 * BDIM * HDIM);

    // ---- Layer 1: U1 = h0 @ W1  (K = 128, N = 384) ----
    {
        const int N = 3 * HDIM, K = HDIM;
        const int waves = (M / 64) * (N / 16);
        gemm_wmma_f32<<<waves / 8, blk256, 0, stream>>>(H0, W1, U, M, N, K);
    }
    sru_scan<<<scanGrid, blk256, 0, stream>>>(
        U, 3 * HDIM, /*res=*/H0, HDIM,
        vc1, b1, c_in + 1 * BDIM * HDIM, H1, cOutBase + 1 * BDIM * HDIM);

    // ---- Layer 2: U2 = h1 @ W2  (K = 128, N = 384) ----
    {
        const int N = 3 * HDIM, K = HDIM;
        const int waves = (M / 64) * (N / 16);
        gemm_wmma_f32<<<waves / 8, blk256, 0, stream>>>(H1, W2, U, M, N, K);
    }
    sru_scan<<<scanGrid, blk256, 0, stream>>>(
        U, 3 * HDIM, /*res=*/H1, HDIM,
        vc2, b2, c_in + 2 * BDIM * HDIM, H0, cOutBase + 2 * BDIM * HDIM);

    // ---- Head: out = h2 @ Wd + bd (one wave per row) ----
    head_gemv<<<M / 8, blk256, 0, stream>>>(H0, Wd, bd, out);
}